// NEG_loss_43843026157952
// MI455X (gfx1250) — compile-verified
//
#include <hip/hip_runtime.h>
#include <hip/hip_bf16.h>
#include <math.h>

// NEG-sampling loss for MI455X (gfx1250, wave32).
// Bandwidth-bound gather workload (~768MB HBM reads, 23.3TB/s -> ~33us floor).
// Dots computed with V_WMMA_F32_16X16X4_F32: one wave handles 16 samples;
// A = EU tile (16x256), 11 B-chains (EV tile + 10 neg tiles), diag(A*B^T)
// gives the 16 per-sample dot products of each chain at zero extra traffic.

typedef __attribute__((ext_vector_type(2))) float v2f;
typedef __attribute__((ext_vector_type(8))) float v8f;

#define DIM 256
#define KSTEPS (DIM / 4)
#define NEGK 10
#define CHAINS (1 + NEGK)
#define TILE 16
#define WAVES_PER_BLOCK 8

__device__ __forceinline__ float log_sigmoid_f(float x) {
  // log(sigmoid(x)) = min(x,0) - log1p(exp(-|x|)), numerically stable
  return fminf(x, 0.0f) - log1pf(expf(-fabsf(x)));
}

// select c[idx] for idx in [0,8) without scratch spills
__device__ __forceinline__ float vsel8(v8f c, int idx) {
  float x = c[0];
  x = (idx == 1) ? c[1] : x;
  x = (idx == 2) ? c[2] : x;
  x = (idx == 3) ? c[3] : x;
  x = (idx == 4) ? c[4] : x;
  x = (idx == 5) ? c[5] : x;
  x = (idx == 6) ? c[6] : x;
  x = (idx == 7) ? c[7] : x;
  return x;
}

__global__ __launch_bounds__(256) void neg_loss_main(
    const int* __restrict__ u, const int* __restrict__ vv,
    const int* __restrict__ negs, const float* __restrict__ embs,
    float* __restrict__ partials, int ntiles) {
  const int lane = threadIdx.x & 31;
  const int tile = blockIdx.x * WAVES_PER_BLOCK + (threadIdx.x >> 5);
  if (tile >= ntiles) return;  // wave-uniform exit; surviving waves keep EXEC all-1s

  const int r = lane & 15;  // row (A: M, B: N) served by this lane
  const int h = lane >> 4;  // 0: K pair {0,1}, 1: K pair {2,3} within each 4-wide K step
  const int s = tile * TILE + r;

  const size_t colh = (size_t)(2 * h);
  const float* pa = embs + (size_t)u[s] * DIM + colh;  // A row: eu[s]
  const float* pb[CHAINS];
  pb[0] = embs + (size_t)vv[s] * DIM + colh;           // chain 0: ev[s]
#pragma unroll
  for (int j = 0; j < NEGK; ++j)
    pb[1 + j] = embs + (size_t)negs[s * NEGK + j] * DIM + colh;  // chains 1..10

  v8f acc[CHAINS];
  const v8f zero = {};
#pragma unroll
  for (int j = 0; j < CHAINS; ++j) acc[j] = zero;

  // K loop: 64 steps of 16x16x4. 12 x b64 loads per step, then 11 WMMAs.
  for (int kk = 0; kk < KSTEPS; ++kk) {
    v2f a = *(const v2f*)(pa + 4 * kk);
    v2f b[CHAINS];
#pragma unroll
    for (int j = 0; j < CHAINS; ++j) b[j] = *(const v2f*)(pb[j] + 4 * kk);
#pragma unroll
    for (int j = 0; j < CHAINS; ++j)
      acc[j] = __builtin_amdgcn_wmma_f32_16x16x4_f32(
          false, a, false, b[j], (short)0, acc[j], false, false);
  }

  // Diagonal of each 16x16 result: sample m<8 -> lane m, vgpr m;
  // sample m>=8 -> lane m+16, vgpr m-8. Active lanes: 0-7 and 24-31, idx = lane&7.
  const bool active = (lane < 8) || (lane >= 24);
  const int idx = lane & 7;
  float lm = 0.0f;
  if (active) {
    lm = log_sigmoid_f(vsel8(acc[0], idx));  // positive pair
#pragma unroll
    for (int j = 1; j < CHAINS; ++j)
      lm += log_sigmoid_f(-vsel8(acc[j], idx));  // noise = -emb[neg] -> score = -dot
  }
  // wave32 sum (inactive lanes contribute 0)
#pragma unroll
  for (int d = 16; d > 0; d >>= 1) lm += __shfl_xor(lm, d, 32);
  if (lane == 0) partials[tile] = lm;
}

__global__ __launch_bounds__(256) void neg_loss_reduce(
    const float* __restrict__ partials, float* __restrict__ out, int ntiles,
    float invN) {
  __shared__ float smem[256];
  float s = 0.0f;
  for (int i = threadIdx.x; i < ntiles; i += 256) s += partials[i];
  smem[threadIdx.x] = s;
  __syncthreads();
#pragma unroll
  for (int stride = 128; stride > 0; stride >>= 1) {
    if ((int)threadIdx.x < stride) smem[threadIdx.x] += smem[threadIdx.x + stride];
    __syncthreads();
  }
  if (threadIdx.x == 0) out[0] = -smem[0] * invN;
}

extern "C" void kernel_launch(void* const* d_in, const int* in_sizes, int n_in,
                              void* d_out, int out_size, void* d_ws, size_t ws_size,
                              hipStream_t stream) {
  const int* u = (const int*)d_in[0];
  const int* v = (const int*)d_in[1];
  const int* negs = (const int*)d_in[2];
  const float* embs = (const float*)d_in[3];
  float* out = (float*)d_out;
  float* partials = (float*)d_ws;  // ntiles floats (16KB for N=65536)

  const int N = in_sizes[0];
  const int ntiles = N / TILE;  // N = 65536 -> 4096 tiles, 1 wave each
  const int blocks = (ntiles + WAVES_PER_BLOCK - 1) / WAVES_PER_BLOCK;

  neg_loss_main<<<blocks, 32 * WAVES_PER_BLOCK, 0, stream>>>(u, v, negs, embs,
                                                             partials, ntiles);
  neg_loss_reduce<<<1, 256, 0, stream>>>(partials, out, ntiles,
                                         1.0f / (float)N);
}